// ESMProjectionHead_4080218931897
// MI455X (gfx1250) — compile-verified
//
#include <hip/hip_runtime.h>
#include <hip/hip_bf16.h>

// ---------------------------------------------------------------------------
// ESM projection head, fp32 WMMA (V_WMMA_F32_16X16X4_F32), gfx1250 wave32.
//   B=64, L=4096, D=128, RL=64, RD=16, DOUT=512
// HBM-bound on the 128MB emb read (~5.5us at 23.3 TB/s); fuse X@B_c with the
// A_c^T contraction so Y[B,L,16] never touches memory. Inner loop reduced to
// one global_load_b64 + one v_wmma per K-step (Bc fragments preloaded to
// registers, mask applied exactly at the Y staging store since mask in {0,1}).
// ---------------------------------------------------------------------------

#define B_SZ   64
#define L_MAX  4096
#define D_DIM  128
#define RL     64
#define RD     16
#define DOUT   512
#define NCH    8                   // L chunks per batch
#define LC     (L_MAX / NCH)       // 512 rows per chunk
#define NWAVE  8
#define TPW    (LC / 16 / NWAVE)   // 4 L-tiles per wave
#define LN_EPS 1e-5f

typedef __attribute__((ext_vector_type(2))) float v2f;
typedef __attribute__((ext_vector_type(8))) float v8f;

// D = A(16x4) * B(4x16) + C, fp32, wave32
__device__ __forceinline__ v8f wmma4(v2f a, v2f b, v8f c) {
  return __builtin_amdgcn_wmma_f32_16x16x4_f32(
      /*neg_a=*/false, a, /*neg_b=*/false, b,
      /*c_mod=*/(short)0, c, /*reuse_a=*/false, /*reuse_b=*/false);
}

// ---------------------------------------------------------------------------
// Kernel 1: per (batch, L-chunk) partial of U = A_c^T * (mask*X @ B_c)
// ---------------------------------------------------------------------------
__global__ __launch_bounds__(256)
void proj_partial_kernel(const float* __restrict__ emb,
                         const unsigned char* __restrict__ mask,
                         const float* __restrict__ Bc,
                         const float* __restrict__ Ac,
                         float* __restrict__ Upart)
{
  const int b    = blockIdx.x;
  const int ch   = blockIdx.y;
  const int tid  = threadIdx.x;
  const int wave = tid >> 5;
  const int lane = tid & 31;
  const int lm   = lane & 15;   // N index / M-in-tile
  const int lh   = lane >> 4;   // lane-half selects K pair / M+8

  __shared__ float Ystage[NWAVE][16 * 16];   // 8 KB: D-layout -> B-layout bounce
  __shared__ float Ublk[NWAVE][RL * RD];     // 32 KB: cross-wave reduction

  // ---- preload all 32 Bc fragments (tile-invariant): bfrag[kb].j = Bc[4kb+j+2lh][lm]
  const float* __restrict__ bcb = Bc + (2 * lh) * RD + lm;
  v2f bfrag[D_DIM / 4];
  #pragma unroll
  for (int kb = 0; kb < D_DIM / 4; ++kb) {
    bfrag[kb].x = bcb[kb * 64];
    bfrag[kb].y = bcb[kb * 64 + 16];
  }

  v8f uacc[4];
  #pragma unroll
  for (int mt = 0; mt < 4; ++mt)
    #pragma unroll
    for (int v = 0; v < 8; ++v) uacc[mt][v] = 0.0f;

  for (int t = 0; t < TPW; ++t) {
    const int l0 = ch * LC + (wave * TPW + t) * 16;
    // X row this lane supplies in A-fragments (+2*lh = starting K column)
    const float* __restrict__ xrow =
        emb + ((size_t)b * L_MAX + (l0 + lm)) * D_DIM + 2 * lh;
    // prefetch next tile's row while this one computes
    if (t + 1 < TPW) __builtin_prefetch(xrow + 16 * D_DIM, 0, 0);

    // ---- Y_tile[16,16] = X_tile[16,128] @ Bc[128,16] : 1 b64 load + 1 wmma/iter
    v8f yacc;
    #pragma unroll
    for (int v = 0; v < 8; ++v) yacc[v] = 0.0f;
    #pragma unroll
    for (int kb = 0; kb < D_DIM / 4; ++kb) {
      const v2f a = *(const v2f*)(xrow + kb * 4);   // global_load_b64, 8B aligned
      yacc = wmma4(a, bfrag[kb], yacc);
    }

    // ---- apply row mask exactly (0.0/1.0) while staging Y to LDS ----
    const unsigned char* __restrict__ mp = mask + (size_t)b * L_MAX + l0 + 8 * lh;
    #pragma unroll
    for (int v = 0; v < 8; ++v)
      Ystage[wave][(v + 8 * lh) * 16 + lm] = yacc[v] * (float)mp[v];

    // ---- preload Ac^T fragments: base + immediate offsets only ----
    // A'[rl, l]: value = Ac[(l0 + kb2*4 + j + 2lh)*RL + mt*16 + lm]
    const float* __restrict__ ab = Ac + ((size_t)(l0 + 2 * lh)) * RL + lm;
    v2f afrag[16];
    #pragma unroll
    for (int kb2 = 0; kb2 < 4; ++kb2)
      #pragma unroll
      for (int mt = 0; mt < 4; ++mt) {
        afrag[kb2 * 4 + mt].x = ab[kb2 * 256 + mt * 16];
        afrag[kb2 * 4 + mt].y = ab[kb2 * 256 + 64 + mt * 16];
      }

    // ---- U += A_c_tile^T[64,16] @ Y_tile[16,16] ----
    #pragma unroll
    for (int kb2 = 0; kb2 < 4; ++kb2) {
      v2f yb;
      yb.x = Ystage[wave][(kb2 * 4 + 2 * lh) * 16 + lm];
      yb.y = Ystage[wave][(kb2 * 4 + 2 * lh + 1) * 16 + lm];
      #pragma unroll
      for (int mt = 0; mt < 4; ++mt)
        uacc[mt] = wmma4(afrag[kb2 * 4 + mt], yb, uacc[mt]);
    }
  }

  // per-wave U -> LDS (flatten rl*16+rd, matching U.reshape order)
  #pragma unroll
  for (int mt = 0; mt < 4; ++mt)
    #pragma unroll
    for (int v = 0; v < 8; ++v)
      Ublk[wave][(mt * 16 + v + 8 * lh) * RD + lm] = uacc[mt][v];
  __syncthreads();

  float* __restrict__ outp = Upart + ((size_t)b * NCH + ch) * (RL * RD);
  for (int i = tid; i < RL * RD; i += 256) {
    float s = 0.0f;
    #pragma unroll
    for (int w = 0; w < NWAVE; ++w) s += Ublk[w][i];
    outp[i] = s;
  }
}

// ---------------------------------------------------------------------------
// Kernel 2: deterministic reduction of the 8 chunk partials -> U[64,1024]
// ---------------------------------------------------------------------------
__global__ __launch_bounds__(256)
void reduce_kernel(const float* __restrict__ Upart, float* __restrict__ Ured)
{
  const int b = blockIdx.x;
  for (int i = threadIdx.x; i < RL * RD; i += 256) {
    float s = 0.0f;
    #pragma unroll
    for (int ch = 0; ch < NCH; ++ch)
      s += Upart[((size_t)b * NCH + ch) * (RL * RD) + i];
    Ured[(size_t)b * (RL * RD) + i] = s;
  }
}

// ---------------------------------------------------------------------------
// Kernel 3: z = U@Hc ; h = z@W1+b1 ; LayerNorm ; ReLU ; out = h@W2+b2
// one block per 16 batch rows; all GEMMs via fp32 WMMA
// ---------------------------------------------------------------------------
__global__ __launch_bounds__(256)
void head_kernel(const float* __restrict__ Ured, const float* __restrict__ Hc,
                 const float* __restrict__ W1,   const float* __restrict__ b1,
                 const float* __restrict__ lnw,  const float* __restrict__ lnb,
                 const float* __restrict__ W2,   const float* __restrict__ b2,
                 float* __restrict__ out)
{
  const int b0   = blockIdx.x * 16;       // batch rows [b0, b0+16)
  const int tid  = threadIdx.x;
  const int wave = tid >> 5;
  const int lane = tid & 31;
  const int lm   = lane & 15;
  const int lh   = lane >> 4;

  __shared__ float S[16 * DOUT];          // 32 KB staging: z, then h
  __shared__ float red[256];
  __shared__ float stat[32];              // [0..15]=mean, [16..31]=inv-std

  // ---- z[16,512] = U[16,1024] @ Hc[1024,512] ----
  v8f zacc[4];
  #pragma unroll
  for (int nt4 = 0; nt4 < 4; ++nt4)
    #pragma unroll
    for (int v = 0; v < 8; ++v) zacc[nt4][v] = 0.0f;

  {
    const float* __restrict__ ub  = Ured + (size_t)(b0 + lm) * (RL * RD) + 2 * lh;
    const float* __restrict__ hcb = Hc + (size_t)(2 * lh) * DOUT + wave * 64 + lm;
    for (int kb = 0; kb < (RL * RD) / 4; ++kb) {
      const v2f a = *(const v2f*)(ub + kb * 4);
      #pragma unroll
      for (int nt4 = 0; nt4 < 4; ++nt4) {
        v2f bb;
        bb.x = hcb[kb * 2048 + nt4 * 16];
        bb.y = hcb[kb * 2048 + 512 + nt4 * 16];
        zacc[nt4] = wmma4(a, bb, zacc[nt4]);
      }
    }
  }
  #pragma unroll
  for (int nt4 = 0; nt4 < 4; ++nt4)
    #pragma unroll
    for (int v = 0; v < 8; ++v)
      S[(v + 8 * lh) * DOUT + (wave * 4 + nt4) * 16 + lm] = zacc[nt4][v];
  __syncthreads();

  // ---- h = z @ W1 + b1 (bias via accumulator init) ----
  v8f hacc[4];
  #pragma unroll
  for (int nt4 = 0; nt4 < 4; ++nt4) {
    const float bias = b1[(wave * 4 + nt4) * 16 + lm];
    #pragma unroll
    for (int v = 0; v < 8; ++v) hacc[nt4][v] = bias;
  }
  {
    const float* __restrict__ w1b = W1 + (size_t)(2 * lh) * DOUT + wave * 64 + lm;
    for (int kb = 0; kb < DOUT / 4; ++kb) {
      const v2f a = *(const v2f*)&S[lm * DOUT + 2 * lh + kb * 4];
      #pragma unroll
      for (int nt4 = 0; nt4 < 4; ++nt4) {
        v2f bb;
        bb.x = w1b[kb * 2048 + nt4 * 16];
        bb.y = w1b[kb * 2048 + 512 + nt4 * 16];
        hacc[nt4] = wmma4(a, bb, hacc[nt4]);
      }
    }
  }
  __syncthreads();                         // everyone done reading z from S
  #pragma unroll
  for (int nt4 = 0; nt4 < 4; ++nt4)
    #pragma unroll
    for (int v = 0; v < 8; ++v)
      S[(v + 8 * lh) * DOUT + (wave * 4 + nt4) * 16 + lm] = hacc[nt4][v];
  __syncthreads();

  // ---- LayerNorm + ReLU over each of the 16 rows (512 cols) ----
  {
    const int r = tid >> 4;                // 16 threads per row
    const int c0 = tid & 15;
    float s = 0.0f;
    for (int j = 0; j < 32; ++j) s += S[r * DOUT + c0 * 32 + j];
    red[tid] = s;
    __syncthreads();
    if (c0 == 0) {
      float m = 0.0f;
      #pragma unroll
      for (int i = 0; i < 16; ++i) m += red[r * 16 + i];
      stat[r] = m * (1.0f / DOUT);
    }
    __syncthreads();
    const float mu = stat[r];
    s = 0.0f;
    for (int j = 0; j < 32; ++j) {
      const float d = S[r * DOUT + c0 * 32 + j] - mu;
      s += d * d;
    }
    red[tid] = s;
    __syncthreads();
    if (c0 == 0) {
      float vv = 0.0f;
      #pragma unroll
      for (int i = 0; i < 16; ++i) vv += red[r * 16 + i];
      stat[16 + r] = rsqrtf(vv * (1.0f / DOUT) + LN_EPS);
    }
    __syncthreads();
    const float inv = stat[16 + r];
    for (int j = 0; j < 32; ++j) {
      const int c = c0 * 32 + j;
      const int idx = r * DOUT + c;
      const float val = (S[idx] - mu) * inv * lnw[c] + lnb[c];
      S[idx] = val > 0.0f ? val : 0.0f;
    }
  }
  __syncthreads();

  // ---- out[16,512] = h @ W2 + b2 ----
  v8f oacc[4];
  #pragma unroll
  for (int nt4 = 0; nt4 < 4; ++nt4) {
    const float bias = b2[(wave * 4 + nt4) * 16 + lm];
    #pragma unroll
    for (int v = 0; v < 8; ++v) oacc[nt4][v] = bias;
  }
  {
    const float* __restrict__ w2b = W2 + (size_t)(2 * lh) * DOUT + wave * 64 + lm;
    for (int kb = 0; kb < DOUT / 4; ++kb) {
      const v2f a = *(const v2f*)&S[lm * DOUT + 2 * lh + kb * 4];
      #pragma unroll
      for (int nt4 = 0; nt4 < 4; ++nt4) {
        v2f bb;
        bb.x = w2b[kb * 2048 + nt4 * 16];
        bb.y = w2b[kb * 2048 + 512 + nt4 * 16];
        oacc[nt4] = wmma4(a, bb, oacc[nt4]);
      }
    }
  }
  #pragma unroll
  for (int nt4 = 0; nt4 < 4; ++nt4)
    #pragma unroll
    for (int v = 0; v < 8; ++v)
      out[(size_t)(b0 + v + 8 * lh) * DOUT + (wave * 4 + nt4) * 16 + lm] =
          oacc[nt4][v];
}

// ---------------------------------------------------------------------------
extern "C" void kernel_launch(void* const* d_in, const int* in_sizes, int n_in,
                              void* d_out, int out_size, void* d_ws, size_t ws_size,
                              hipStream_t stream) {
  const float*         emb  = (const float*)d_in[0];
  const unsigned char* mask = (const unsigned char*)d_in[1];  // jnp bool
  const float* Bc  = (const float*)d_in[2];
  const float* Ac  = (const float*)d_in[3];
  const float* Hc  = (const float*)d_in[4];
  const float* W1  = (const float*)d_in[5];
  const float* b1  = (const float*)d_in[6];
  const float* lnw = (const float*)d_in[7];
  const float* lnb = (const float*)d_in[8];
  const float* W2  = (const float*)d_in[9];
  const float* b2  = (const float*)d_in[10];
  float* out = (float*)d_out;

  float* Upart = (float*)d_ws;                         // [64][8][1024] = 2 MB
  float* UredG = Upart + (size_t)B_SZ * NCH * RL * RD; // [64][1024]

  proj_partial_kernel<<<dim3(B_SZ, NCH), 256, 0, stream>>>(emb, mask, Bc, Ac, Upart);
  reduce_kernel<<<B_SZ, 256, 0, stream>>>(Upart, UredG);
  head_kernel<<<B_SZ / 16, 256, 0, stream>>>(UredG, Hc, W1, b1, lnw, lnb, W2, b2, out);
}